// SOTLayer_40123584479808
// MI455X (gfx1250) — compile-verified
//
#include <hip/hip_runtime.h>
#include <hip/hip_bf16.h>
#include <stdint.h>

// SOT layer: binary-tree descent (17 levels) + streaming node update.
// Descent: latency-bound -> single wave32, b128 loads, 2-level-deep
//          global_prefetch_b8 speculation to hide the dependent-load chain.
// Update:  268 MB one-pass stream @ 23.3 TB/s (~11.5us) -> nontemporal b128
//          load/store, 8 rows in flight per wave, X broadcast staged into LDS
//          with the CDNA5 async-to-LDS engine (ASYNCcnt).

#define TDEPTH 17
#define DIM    128
#define NNODES ((2 << TDEPTH) - 1)   // 262143

typedef float vf4 __attribute__((ext_vector_type(4)));

__global__ __launch_bounds__(32)
void sot_descend(const float* __restrict__ X,
                 const float* __restrict__ nodes,
                 float* __restrict__ out_bmu,   // d_out[0]
                 int*   __restrict__ ws_bmu) {  // handoff to update kernel
    const int lane = threadIdx.x;                       // wave32: 4 dims/lane
    const vf4 x4 = *reinterpret_cast<const vf4*>(X + lane * 4);

    int g = 0;                                          // current BMU (root)
    #pragma unroll 1
    for (int level = 1; level <= TDEPTH; ++level) {
        const size_t c1 = 2 * (size_t)g + 1;
        const size_t c2 = c1 + 1;

        // Speculative prefetch: 4 grandchildren (2KB, 64B/lane) and
        // 8 great-grandchildren (4KB, 128B/lane). Decision latency per level
        // then overlaps with the next two levels' memory traffic.
        if (level < TDEPTH) {
            const char* gp = (const char*)(nodes + (4 * (size_t)g + 3) * DIM);
            __builtin_prefetch(gp + lane * 64, 0, 0);   // global_prefetch_b8
        }
        if (level < TDEPTH - 1) {
            const char* gp2 = (const char*)(nodes + (8 * (size_t)g + 7) * DIM);
            __builtin_prefetch(gp2 + lane * 128, 0, 0);
        }

        const vf4 a = *reinterpret_cast<const vf4*>(nodes + c1 * DIM + lane * 4);
        const vf4 b = *reinterpret_cast<const vf4*>(nodes + c2 * DIM + lane * 4);

        const vf4 da = a - x4;
        const vf4 db = b - x4;
        float d1 = da.x * da.x + da.y * da.y + da.z * da.z + da.w * da.w;
        float d2 = db.x * db.x + db.y * db.y + db.z * db.z + db.w * db.w;

        // wave32 butterfly reduction over 128 dims
        #pragma unroll
        for (int m = 16; m; m >>= 1) {
            d1 += __shfl_xor(d1, m, 32);
            d2 += __shfl_xor(d2, m, 32);
        }
        // reference: argmin over [d1,d2]; outer ^2 is monotonic -> compare
        // sums. tie -> first element (c1), matching jnp.argmin.
        g = (d2 < d1) ? (int)c2 : (int)c1;
    }
    if (lane == 0) {
        out_bmu[0] = (float)g;   // tuple output slot 0 (concatenated as f32)
        ws_bmu[0]  = g;          // leaf index encodes the full BMU path
    }
}

#define ROWS_PER_BLOCK 64        // 8 waves x 8 rows each

__global__ __launch_bounds__(256)
void sot_update(const float* __restrict__ X,
                const float* __restrict__ nodes,
                const int*   __restrict__ ws_bmu,
                float*       __restrict__ out_nodes) {  // d_out + 1
    __shared__ float xs[DIM];

    // Stage the 512B X vector into LDS with the gfx1250 async-to-LDS path:
    // wave 0 (lanes 0..31) each moves one b128; tracked by ASYNCcnt.
    if (threadIdx.x < 32) {
        uint32_t lds = (uint32_t)(uintptr_t)(&xs[threadIdx.x * 4]);
        uint64_t ga  = (uint64_t)(uintptr_t)(X + threadIdx.x * 4);
        asm volatile("global_load_async_to_lds_b128 %0, %1, off"
                     :: "v"(lds), "v"(ga) : "memory");
        asm volatile("s_wait_asynccnt 0" ::: "memory");
    }
    __syncthreads();

    const int lane = threadIdx.x & 31;
    const int wave = threadIdx.x >> 5;
    const int bmu1 = ws_bmu[0] + 1;              // 18-bit path, MSB = bit 17
    const vf4 xv = *reinterpret_cast<const vf4*>(&xs[lane * 4]);

    // Each wave owns 8 independent 512B rows -> 8 b128 loads in flight.
    const int base = blockIdx.x * ROWS_PER_BLOCK + wave;
    #pragma unroll
    for (int i = 0; i < 8; ++i) {
        const int node = base + i * 8;
        if (node < NNODES) {
            // layer_state = common-prefix length of node's heap path vs the
            // BMU leaf's path; lr = LR * 2^(state+1) / 2^(DEPTH+1).
            float lr = 0.0f;
            if (node > 0) {
                const int a    = node + 1;
                const int n    = 31 - __clz(a);          // tree level
                const int btop = bmu1 >> (TDEPTH - n);
                const int x    = a ^ btop;
                const int st   = x ? (n + __clz(x) - 32) : n;
                lr = ldexpf(0.3f, st - TDEPTH);          // one v_ldexp_f32
            }
            const size_t off = (size_t)node * DIM + (size_t)lane * 4;
            // one-pass stream: non-temporal so 268MB doesn't rinse L2
            const vf4 v = __builtin_nontemporal_load(
                reinterpret_cast<const vf4*>(nodes + off));
            const vf4 o = v + lr * (xv - v);
            __builtin_nontemporal_store(
                o, reinterpret_cast<vf4*>(out_nodes + off));
        }
    }
}

extern "C" void kernel_launch(void* const* d_in, const int* in_sizes, int n_in,
                              void* d_out, int out_size, void* d_ws, size_t ws_size,
                              hipStream_t stream) {
    const float* X     = (const float*)d_in[0];   // (128,)
    const float* nodes = (const float*)d_in[1];   // (262143, 128)
    float* out = (float*)d_out;                   // [bmu | new_nodes]
    int*   ws  = (int*)d_ws;

    sot_descend<<<1, 32, 0, stream>>>(X, nodes, out, ws);

    const int nblocks = (NNODES + ROWS_PER_BLOCK - 1) / ROWS_PER_BLOCK; // 4096
    sot_update<<<nblocks, 256, 0, stream>>>(X, nodes, ws, out + 1);
}